// TMSCNNTexturesToFlattenedGroups_74294344286542
// MI455X (gfx1250) — compile-verified
//
#include <hip/hip_runtime.h>

// Pure gather: out[b, gk, :] = textures[b, group_idx[gk], :]   (C = 16 floats = 64B rows)
// Roofline: ~576MB NT output stream + ~18MB NT index stream from HBM; the 128MB
// texture set stays L2-resident (192MB L2) because only the gathered reads use the
// default RT temporal policy. Lower bound ~31us at 23.3 TB/s.

typedef float v4f __attribute__((ext_vector_type(4)));

template <int STATIC_B>
__global__ __launch_bounds__(256)
void TMSCNNTexturesToFlattenedGroups_74294344286542_kernel(
    const float* __restrict__ textures,   // [B, T, C] flat
    const int*   __restrict__ group_idx,  // [G*K] flat
    float*       __restrict__ out,        // [B, G*K, C] flat
    long long GK,                         // G*K
    long long TC,                         // T*C (texture stride per batch, floats)
    int runtimeB)
{
    constexpr int C   = 16;      // channels per texel
    constexpr int VEC = 4;       // float4 chunk
    constexpr int CV  = C / VEC; // 4 chunks per 64B row

    const long long tid   = (long long)blockIdx.x * blockDim.x + threadIdx.x;
    const long long total = GK * CV;
    if (tid >= total) return;

    const int       chunk = (int)(tid & (CV - 1));   // 0..3 within a 64B row
    const long long gk    = tid >> 2;                // which (g,k) entry

    // Index stream: read once, non-temporal (4 adjacent lanes coalesce on one dword).
    const int idx = __builtin_nontemporal_load(&group_idx[gk]);

    const long long texel_off   = (long long)idx * C + (long long)chunk * VEC;
    const long long out_off     = gk * C + (long long)chunk * VEC;
    const long long out_bstride = GK * C;

    if constexpr (STATIC_B > 0) {
        // Fully unrolled straight-line path (no loop control at all).
        #pragma unroll
        for (int b = 0; b < STATIC_B; ++b) {
            const v4f v = *(const v4f*)(textures + (long long)b * TC + texel_off);
            __builtin_nontemporal_store(v, (v4f*)(out + (long long)b * out_bstride + out_off));
        }
    } else {
        // Generic fallback for arbitrary runtime batch count.
        for (int b = 0; b < runtimeB; ++b) {
            const v4f v = *(const v4f*)(textures + (long long)b * TC + texel_off);
            __builtin_nontemporal_store(v, (v4f*)(out + (long long)b * out_bstride + out_off));
        }
    }
}

extern "C" void kernel_launch(void* const* d_in, const int* in_sizes, int n_in,
                              void* d_out, int out_size, void* d_ws, size_t ws_size,
                              hipStream_t stream) {
    // setup_inputs order: mesh_ids [B] (unused by reference), textures [B,T,C], group_idx [G,K]
    const float* textures  = (const float*)d_in[1];
    const int*   group_idx = (const int*)d_in[2];
    float*       out       = (float*)d_out;

    const int       B  = in_sizes[0];                 // batch (mesh_ids length)
    const long long GK = (long long)in_sizes[2];      // G*K flattened
    const long long TC = (long long)in_sizes[1] / B;  // T*C floats per batch

    const long long total = GK * 4;                   // one thread per float4 chunk
    const int       block = 256;                      // 8 wave32s per block
    const unsigned  grid  = (unsigned)((total + block - 1) / block);

    switch (B) {
    case 1:
        TMSCNNTexturesToFlattenedGroups_74294344286542_kernel<1>
            <<<grid, block, 0, stream>>>(textures, group_idx, out, GK, TC, B);
        break;
    case 2:
        TMSCNNTexturesToFlattenedGroups_74294344286542_kernel<2>
            <<<grid, block, 0, stream>>>(textures, group_idx, out, GK, TC, B);
        break;
    case 4:
        TMSCNNTexturesToFlattenedGroups_74294344286542_kernel<4>
            <<<grid, block, 0, stream>>>(textures, group_idx, out, GK, TC, B);
        break;
    default:
        TMSCNNTexturesToFlattenedGroups_74294344286542_kernel<0>
            <<<grid, block, 0, stream>>>(textures, group_idx, out, GK, TC, B);
        break;
    }
}